// NormalizedCenterLoss_56521769615884
// MI455X (gfx1250) — compile-verified
//
#include <hip/hip_runtime.h>

// NormalizedCenterLoss on gfx1250 (MI455X), wave32.
//
//   loss = (1/N) * sum_i ( 1 - dot(x_i, c_{l_i}) / (max(|x_i|,eps)*max(|c_{l_i}|,eps)) )
//
// The reference's NxC matmul is masked down to one column per row, so we only
// gather 4096 center rows (~4MB total traffic -> ~0.2us at 23.3 TB/s).
// Dot products + squared norms are computed on the WMMA pipe via
// V_WMMA_F32_16X16X4_F32 accumulation chains (diagonals of X*C^T, X*X^T, C*C^T).

typedef float v2f __attribute__((ext_vector_type(2)));
typedef float v8f __attribute__((ext_vector_type(8)));

#define FEAT   128
#define TILE   16
#define LDSPAD 4                 // row stride 132 floats -> conflict-free frag reads
#define EPSN   1e-12f

__global__ __launch_bounds__(32)
void center_loss_tile_kernel(const float* __restrict__ x,
                             const long long* __restrict__ labels,
                             const float* __restrict__ centers,
                             float* __restrict__ partial,
                             int batch)
{
    __shared__ float xs[TILE][FEAT + LDSPAD];
    __shared__ float cs[TILE][FEAT + LDSPAD];
    __shared__ float dxc[TILE];
    __shared__ float dxx[TILE];
    __shared__ float dcc[TILE];

    const int lane  = threadIdx.x;          // 0..31, one wave per block
    const int tile0 = blockIdx.x * TILE;

    // ---- Stage 1: cooperative gather into LDS (float4, coalesced per row) ----
    for (int t = lane; t < TILE * (FEAT / 4); t += 32) {
        const int row = t >> 5;             // FEAT/4 == 32 float4 per row
        const int c4  = t & 31;
        const int srow = tile0 + row;
        const int xsrc = (srow < batch) ? srow : 0;
        const long long lbl = labels[xsrc];
        const float4 xv = ((const float4*)(x       + (size_t)xsrc * FEAT))[c4];
        const float4 cv = ((const float4*)(centers + (size_t)lbl  * FEAT))[c4];
        *(float4*)&xs[row][c4 * 4] = xv;
        *(float4*)&cs[row][c4 * 4] = cv;
    }
    __syncthreads();

    // ---- Stage 2: WMMA accumulation chains over K = 128 in steps of 4 ----
    // A-frag layout (ISA 7.12.2, 32-bit A 16x4): lanes 0-15 hold M=lane,
    // K=(k,k+1); lanes 16-31 hold M=lane-16, K=(k+2,k+3).  B 4x16 mirrors it
    // with N=lane&15, so the *same* per-lane indexing builds both operands.
    const int m    = lane & 15;
    const int koff = (lane >> 4) * 2;

    v8f axc = {};   // diag -> dot(x_i, c_i)
    v8f axx = {};   // diag -> |x_i|^2
    v8f acc = {};   // diag -> |c_i|^2

    #pragma unroll
    for (int k = 0; k < FEAT; k += 4) {
        const v2f fx = *(const v2f*)&xs[m][k + koff];
        const v2f fc = *(const v2f*)&cs[m][k + koff];
        axc = __builtin_amdgcn_wmma_f32_16x16x4_f32(false, fx, false, fc,
                                                    (short)0, axc, false, false);
        axx = __builtin_amdgcn_wmma_f32_16x16x4_f32(false, fx, false, fx,
                                                    (short)0, axx, false, false);
        acc = __builtin_amdgcn_wmma_f32_16x16x4_f32(false, fc, false, fc,
                                                    (short)0, acc, false, false);
    }

    // ---- Stage 3: pull diagonals out of the 16x16 accumulators ----
    // C/D layout: VGPR v, lane l<16 -> (M=v, N=l); lane l>=16 -> (M=v+8, N=l-16).
    // Diagonal i=v lives in lane v; i=v+8 lives in lane v+24.
    #pragma unroll
    for (int v = 0; v < 8; ++v) {
        if (lane == v)      { dxc[v]     = axc[v]; dxx[v]     = axx[v]; dcc[v]     = acc[v]; }
        if (lane == v + 24) { dxc[v + 8] = axc[v]; dxx[v + 8] = axx[v]; dcc[v + 8] = acc[v]; }
    }
    __syncthreads();

    // ---- Stage 4: per-sample loss + wave32 reduction ----
    float contrib = 0.0f;
    if (lane < TILE) {
        const int s = tile0 + lane;
        if (s < batch) {
            const float nx = fmaxf(sqrtf(dxx[lane]), EPSN);
            const float nc = fmaxf(sqrtf(dcc[lane]), EPSN);
            contrib = 1.0f - dxc[lane] / (nx * nc);
        }
    }
    #pragma unroll
    for (int off = 16; off > 0; off >>= 1)
        contrib += __shfl_down(contrib, off, 32);
    if (lane == 0)
        partial[blockIdx.x] = contrib;
}

__global__ __launch_bounds__(256)
void center_loss_reduce_kernel(const float* __restrict__ partial, int n,
                               float* __restrict__ out, float inv_batch)
{
    __shared__ float sm[8];
    float v = 0.0f;
    for (int i = threadIdx.x; i < n; i += 256)
        v += partial[i];
    #pragma unroll
    for (int off = 16; off > 0; off >>= 1)
        v += __shfl_down(v, off, 32);
    if ((threadIdx.x & 31) == 0)
        sm[threadIdx.x >> 5] = v;
    __syncthreads();
    if (threadIdx.x < 8) {
        v = sm[threadIdx.x];
        #pragma unroll
        for (int off = 4; off > 0; off >>= 1)
            v += __shfl_down(v, off, 32);
        if (threadIdx.x == 0)
            out[0] = v * inv_batch;
    }
}

extern "C" void kernel_launch(void* const* d_in, const int* in_sizes, int n_in,
                              void* d_out, int out_size, void* d_ws, size_t ws_size,
                              hipStream_t stream)
{
    const float*     x       = (const float*)d_in[0];      // [batch, 128] f32
    const long long* labels  = (const long long*)d_in[1];  // [batch] i64
    const float*     centers = (const float*)d_in[2];      // [C, 128] f32

    const int batch  = in_sizes[1];
    const int ntiles = (batch + TILE - 1) / TILE;

    float* partial = (float*)d_ws;                         // ntiles floats of scratch

    center_loss_tile_kernel<<<ntiles, 32, 0, stream>>>(x, labels, centers,
                                                       partial, batch);
    center_loss_reduce_kernel<<<1, 256, 0, stream>>>(partial, ntiles,
                                                     (float*)d_out,
                                                     1.0f / (float)batch);
}